// ConnectionTopology_3186865734121
// MI455X (gfx1250) — compile-verified
//
#include <hip/hip_runtime.h>
#include <stdint.h>

#define PROTOS 4096
#define BATCH  4096
#define AGE_LIMIT 50.0f

typedef unsigned int u32x4 __attribute__((ext_vector_type(4)));
typedef int          i32x8 __attribute__((ext_vector_type(8)));
typedef int          i32x4 __attribute__((ext_vector_type(4)));

// ---------------------------------------------------------------------------
// TDM descriptor for one matrix row (4096 f32 = 16 KiB), 1-D tile.
// D# group0: [1:0]=count=1, [63:32]=lds_addr, [120:64]=global_addr, [127:126]=type=2
// D# group1: [17:16]=data_size(2 -> 4B), [79:48]=tensor_dim0, [111:80]=tensor_dim1,
//            [127:112]=tile_dim0, [207:160]=tensor_dim0_stride
// Toolchain here is the 6-arg builtin form (clang-23 / therock-10.0 headers).
// ---------------------------------------------------------------------------
__device__ __forceinline__ void tdm_row(bool is_store, unsigned lds_off, const float* gaddr) {
  unsigned long long ga = (unsigned long long)(uintptr_t)gaddr;
  u32x4 g0;
  g0.x = 1u;                                                  // count=1 (user descriptor)
  g0.y = lds_off;                                             // lds_addr (bytes)
  g0.z = (unsigned)(ga & 0xFFFFFFFFull);                      // global_addr[31:0]
  g0.w = (unsigned)((ga >> 32) & 0x1FFFFFFull) | (2u << 30);  // global_addr[56:32], type=2
  i32x8 g1;
  g1[0] = (int)(2u << 16);                  // workgroup_mask=0, data_size=4B
  g1[1] = (int)((unsigned)PROTOS << 16);    // tensor_dim0[15:0] in bits 63:48
  g1[2] = (int)(1u << 16);                  // tensor_dim0[31:16]=0, tensor_dim1=1
  g1[3] = (int)((unsigned)PROTOS << 16);    // tensor_dim1[31:16]=0, tile_dim0=4096
  g1[4] = 0;                                // tile_dim1=0, tile_dim2=0 (unused)
  g1[5] = PROTOS;                           // tensor_dim0_stride[31:0]
  g1[6] = 0;                                // stride hi, tensor_dim1_stride lo
  g1[7] = 0;
  i32x4 g2 = {0, 0, 0, 0};
  i32x4 g3 = {0, 0, 0, 0};
  i32x8 g4 = {0, 0, 0, 0, 0, 0, 0, 0};      // extra operand of the 6-arg form
  if (is_store) __builtin_amdgcn_tensor_store_from_lds(g0, g1, g2, g3, g4, 0);
  else          __builtin_amdgcn_tensor_load_to_lds(g0, g1, g2, g3, g4, 0);
}

// ---------------------------------------------------------------------------
// Phase 1: stable top-2 minima per row of d.  One wave32 per row.
// ---------------------------------------------------------------------------
__device__ __forceinline__ bool lex_less(float va, int ia, float vb, int ib) {
  return (va < vb) || ((va == vb) && (ia < ib));   // argsort-stable tie break
}

__global__ void __launch_bounds__(256)
top2_rows_kernel(const float* __restrict__ d, int2* __restrict__ pairs) {
  const int lane = threadIdx.x & 31;
  const int wave = threadIdx.x >> 5;
  const int row  = blockIdx.x * 8 + wave;
  if (row >= BATCH) return;

  const float4* rp = (const float4*)(d + (size_t)row * PROTOS);
  float v0 = __builtin_inff(), v1 = __builtin_inff();
  int   b0 = 0x7fffffff,       b1 = 0x7fffffff;

  for (int c = lane; c < PROTOS / 4; c += 32) {
    __builtin_prefetch(rp + c + 64, 0, 0);     // global_prefetch_b8
    float4 x = rp[c];
    const int j = c * 4;
    float xv[4] = {x.x, x.y, x.z, x.w};
#pragma unroll
    for (int k = 0; k < 4; ++k) {
      float v = xv[k]; int idx = j + k;
      if (lex_less(v, idx, v0, b0))      { v1 = v0; b1 = b0; v0 = v; b0 = idx; }
      else if (lex_less(v, idx, v1, b1)) { v1 = v;  b1 = idx; }
    }
  }
  // wave32 butterfly merge of (val,idx) top-2 sets
#pragma unroll
  for (int m = 16; m >= 1; m >>= 1) {
    float ov0 = __shfl_xor(v0, m, 32); int ob0 = __shfl_xor(b0, m, 32);
    float ov1 = __shfl_xor(v1, m, 32); int ob1 = __shfl_xor(b1, m, 32);
    if (lex_less(ov0, ob0, v0, b0))      { v1 = v0;  b1 = b0;  v0 = ov0; b0 = ob0; }
    else if (lex_less(ov0, ob0, v1, b1)) { v1 = ov0; b1 = ob0; }
    if (lex_less(ov1, ob1, v0, b0))      { v1 = v0;  b1 = b0;  v0 = ov1; b0 = ob1; }
    else if (lex_less(ov1, ob1, v1, b1)) { v1 = ov1; b1 = ob1; }
  }
  if (lane == 0) pairs[row] = make_int2(b0, b1);
}

// ---------------------------------------------------------------------------
// Phase 2: sequential edge/age scan. One workgroup; wave 0 drives the TDM to
// DMA the 4 touched rows into LDS, all 1024 threads apply the fused update,
// TDM stores them back. TDM ops from one wave are in-order; s_wait_tensorcnt
// plus workgroup barriers fence LDS reuse across steps.
// ---------------------------------------------------------------------------
__global__ void __launch_bounds__(1024)
topology_scan_kernel(const int2* __restrict__ pairs, float* __restrict__ out) {
  __shared__ float lds[4 * PROTOS];                 // cm0 | cm1 | ag0 | ag1 = 64 KiB
  float* cmat = out;
  float* age  = out + (size_t)PROTOS * PROTOS;
  const unsigned base    = (unsigned)(uintptr_t)(&lds[0]);   // low 32 bits = LDS offset
  const unsigned off_cm0 = base;
  const unsigned off_cm1 = base + 1u * PROTOS * 4u;
  const unsigned off_ag0 = base + 2u * PROTOS * 4u;
  const unsigned off_ag1 = base + 3u * PROTOS * 4u;
  const int tid = threadIdx.x;

  for (int s = 0; s < BATCH; ++s) {
    int2 pr = pairs[s];
    const int i0 = __builtin_amdgcn_readfirstlane(pr.x);
    const int i1 = __builtin_amdgcn_readfirstlane(pr.y);

    if (tid < 32) {                                  // wave 0 issues the DMA
      tdm_row(false, off_cm0, cmat + (size_t)i0 * PROTOS);
      tdm_row(false, off_cm1, cmat + (size_t)i1 * PROTOS);
      tdm_row(false, off_ag0, age  + (size_t)i0 * PROTOS);
      tdm_row(false, off_ag1, age  + (size_t)i1 * PROTOS);
      __builtin_amdgcn_s_wait_tensorcnt(0);
    }
    __syncthreads();

#pragma unroll
    for (int k = 0; k < PROTOS / 1024; ++k) {
      const int j = tid + k * 1024;
      // row i0: set edge (i0,i1)+reset age, age live edges, prune > AGE_LIMIT
      float c = lds[j], a = lds[2 * PROTOS + j];
      if (j == i1) { c = 1.0f; a = 0.0f; }
      if (c == 1.0f) a += 1.0f;
      if (a > AGE_LIMIT) c = 0.0f;
      lds[j] = c; lds[2 * PROTOS + j] = a;
      // row i1 (mirror)
      c = lds[PROTOS + j]; a = lds[3 * PROTOS + j];
      if (j == i0) { c = 1.0f; a = 0.0f; }
      if (c == 1.0f) a += 1.0f;
      if (a > AGE_LIMIT) c = 0.0f;
      lds[PROTOS + j] = c; lds[3 * PROTOS + j] = a;
    }
    __syncthreads();

    if (tid < 32) {                                  // DMA rows back
      tdm_row(true, off_cm0, cmat + (size_t)i0 * PROTOS);
      tdm_row(true, off_cm1, cmat + (size_t)i1 * PROTOS);
      tdm_row(true, off_ag0, age  + (size_t)i0 * PROTOS);
      tdm_row(true, off_ag1, age  + (size_t)i1 * PROTOS);
      __builtin_amdgcn_s_wait_tensorcnt(0);
    }
    __syncthreads();
  }
}

// ---------------------------------------------------------------------------
extern "C" void kernel_launch(void* const* d_in, const int* in_sizes, int n_in,
                              void* d_out, int out_size, void* d_ws, size_t ws_size,
                              hipStream_t stream) {
  (void)in_sizes; (void)n_in; (void)out_size; (void)ws_size;
  const float* d     = (const float*)d_in[0];
  const float* cmat0 = (const float*)d_in[1];
  const float* age0  = (const float*)d_in[2];
  float* out  = (float*)d_out;
  int2* pairs = (int2*)d_ws;

  const size_t mat_bytes = (size_t)PROTOS * PROTOS * sizeof(float);
  (void)hipMemcpyAsync(out, cmat0, mat_bytes, hipMemcpyDeviceToDevice, stream);
  (void)hipMemcpyAsync(out + (size_t)PROTOS * PROTOS, age0, mat_bytes,
                       hipMemcpyDeviceToDevice, stream);

  top2_rows_kernel<<<BATCH / 8, 256, 0, stream>>>(d, pairs);
  topology_scan_kernel<<<1, 1024, 0, stream>>>(pairs, out);
}